// EMLTree_18210661335269
// MI455X (gfx1250) — compile-verified
//
#include <hip/hip_runtime.h>
#include <math.h>

// ---------------------------------------------------------------------------
// EML tree over complex128, N = 4M elements.
// Compute-bound: 14 log + 14 atan2 + 7 exp + 7 sincos (fp64) per element.
// Data path: async global->LDS double buffering (CDNA5 ASYNCcnt engine).
// ---------------------------------------------------------------------------

struct Cx { double re, im; };

// eml(a,b) = exp(log(a) * log(b)), complex128.
// log(z) = 0.5*log(re^2+im^2) + i*atan2(im,re); exp(z) = e^re * (cos im, sin im)
__device__ __forceinline__ Cx eml_c(const Cx a, const Cx b) {
    double lar = 0.5 * log(a.re * a.re + a.im * a.im);
    double lai = atan2(a.im, a.re);
    double lbr = 0.5 * log(b.re * b.re + b.im * b.im);
    double lbi = atan2(b.im, b.re);
    double pr = lar * lbr - lai * lbi;
    double pi = lar * lbi + lai * lbr;
    double e = exp(pr);
    double s, c;
    sincos(pi, &s, &c);
    Cx r; r.re = e * c; r.im = e * s;
    return r;
}

// Pre-pass: softmax of (8x2) leaf logits and (6x3) mixing logits -> 34 doubles
// in workspace. Done once so the up-to-1M-thread main kernel reads plain (uniform,
// scalar-loadable) weights instead of each thread redoing 34 fp64 exps.
__global__ void eml_weights_kernel(const double* __restrict__ leaf,
                                   const double* __restrict__ mix,
                                   double* __restrict__ w) {
    if (blockIdx.x == 0 && threadIdx.x == 0) {
        #pragma unroll
        for (int i = 0; i < 8; ++i) {
            double a = leaf[2 * i], b = leaf[2 * i + 1];
            double m = fmax(a, b);
            double ea = exp(a - m), eb = exp(b - m);
            double s = ea + eb;
            w[2 * i] = ea / s;
            w[2 * i + 1] = eb / s;
        }
        #pragma unroll
        for (int i = 0; i < 6; ++i) {
            double a = mix[3 * i], b = mix[3 * i + 1], c = mix[3 * i + 2];
            double m = fmax(a, fmax(b, c));
            double ea = exp(a - m), eb = exp(b - m), ec = exp(c - m);
            double s = ea + eb + ec;
            w[16 + 3 * i] = ea / s;
            w[16 + 3 * i + 1] = eb / s;
            w[16 + 3 * i + 2] = ec / s;
        }
    }
}

// Main kernel: 256 threads = 8 wave32. Each wave async-stages its 32 complex128
// x values (512 B) into a private double-buffered LDS tile, so no VGPRs sit on
// load latency; compute reads via ds_load_b128.
__global__ void __launch_bounds__(256)
eml_tree_kernel(const double2* __restrict__ x,
                const double* __restrict__ w,
                double2* __restrict__ out,
                long n) {
    extern __shared__ double2 sbuf[];   // dynamic LDS only -> starts at offset 0

    const unsigned lane  = threadIdx.x & 31u;
    const unsigned wv    = threadIdx.x >> 5;
    // Byte addresses inside LDS for this lane's slot, buffers 0/1.
    const unsigned base0 = wv * 1024u + lane * 16u;
    const unsigned base1 = base0 + 512u;
    // double2 element index of buffer-0 slot.
    const unsigned eidx0 = wv * 64u + lane;

    long g = (long)blockIdx.x * blockDim.x + threadIdx.x;
    const long stride = (long)gridDim.x * (long)blockDim.x;

    // Uniform weight loads (pointer + literal indices -> scalar loads).
    double lw[8][2];
    double mw[6][3];
    #pragma unroll
    for (int i = 0; i < 8; ++i) {
        lw[i][0] = w[2 * i];
        lw[i][1] = w[2 * i + 1];
    }
    #pragma unroll
    for (int i = 0; i < 6; ++i) {
        mw[i][0] = w[16 + 3 * i];
        mw[i][1] = w[16 + 3 * i + 1];
        mw[i][2] = w[16 + 3 * i + 2];
    }

    // Prologue: stage first tile into buffer 0 (ASYNCcnt path).
    if (g < n) {
        const double2* gp = x + g;
        asm volatile("global_load_async_to_lds_b128 %0, %1, off"
                     :: "v"(base0), "v"(gp) : "memory");
    }

    int buf = 0;
    while (g < n) {
        long gn = g + stride;
        if (gn < n) {
            // Prefetch next tile into the other buffer, then wait until only
            // that one is outstanding (current tile has landed in LDS).
            unsigned nb = buf ? base0 : base1;
            const double2* gp = x + gn;
            asm volatile("global_load_async_to_lds_b128 %0, %1, off"
                         :: "v"(nb), "v"(gp) : "memory");
            asm volatile("s_wait_asynccnt 1" ::: "memory");
        } else {
            asm volatile("s_wait_asynccnt 0" ::: "memory");
        }

        double2 xv = sbuf[eidx0 + (buf ? 32u : 0u)];   // ds_load_b128
        const double xr = xv.x, xi = xv.y;

        // Leaves: real weights -> vals_i = (lw0 + lw1*xr, lw1*xi)
        Cx v[8];
        #pragma unroll
        for (int i = 0; i < 8; ++i) {
            v[i].re = lw[i][0] + lw[i][1] * xr;
            v[i].im = lw[i][1] * xi;
        }

        // Level 3 -> 4 nodes, mixing rows 2..5
        Cx u[4];
        #pragma unroll
        for (int j = 0; j < 4; ++j) {
            Cx c = eml_c(v[2 * j], v[2 * j + 1]);
            const double w0 = mw[2 + j][0], w1 = mw[2 + j][1], w2 = mw[2 + j][2];
            u[j].re = w0 + w1 * xr + w2 * c.re;
            u[j].im = w1 * xi + w2 * c.im;
        }

        // Level 2 -> 2 nodes, mixing rows 0..1
        Cx t[2];
        #pragma unroll
        for (int j = 0; j < 2; ++j) {
            Cx c = eml_c(u[2 * j], u[2 * j + 1]);
            const double w0 = mw[j][0], w1 = mw[j][1], w2 = mw[j][2];
            t[j].re = w0 + w1 * xr + w2 * c.re;
            t[j].im = w1 * xi + w2 * c.im;
        }

        // Root
        Cx o = eml_c(t[0], t[1]);
        double2 ov;
        ov.x = o.re;
        ov.y = o.im;
        out[g] = ov;                                   // global_store_b128

        buf ^= 1;
        g = gn;
    }
}

extern "C" void kernel_launch(void* const* d_in, const int* in_sizes, int n_in,
                              void* d_out, int out_size, void* d_ws, size_t ws_size,
                              hipStream_t stream) {
    (void)n_in; (void)out_size; (void)ws_size;

    const double2* x    = (const double2*)d_in[0];   // complex128 interleaved
    const double*  leaf = (const double*)d_in[1];    // (8,2) float64
    const double*  mix  = (const double*)d_in[2];    // (6,3) float64
    double2* out = (double2*)d_out;                  // complex128 interleaved
    double*  w   = (double*)d_ws;                    // 34 softmax weights

    long n = (long)in_sizes[0];

    hipLaunchKernelGGL(eml_weights_kernel, dim3(1), dim3(32), 0, stream,
                       leaf, mix, w);

    long blocks = (n + 255) / 256;
    if (blocks > 4096) blocks = 4096;   // grid-stride; ~1M threads fill the chip
    if (blocks < 1) blocks = 1;

    // 8 waves * 2 buffers * 512 B = 8 KiB dynamic LDS per block
    hipLaunchKernelGGL(eml_tree_kernel, dim3((unsigned)blocks), dim3(256), 8192,
                       stream, x, w, out, n);
}